// Mesh1_61667140436413
// MI455X (gfx1250) — compile-verified
//
#include <hip/hip_runtime.h>

// ---------------------------------------------------------------------------
// Mesh1: 4 chained GEMVs + concat + neighbour-gather. Memory-bound (~115 MB
// of weights, 0.5 FLOP/byte -> HBM roofline ~5us @ 23.3 TB/s; weights fit in
// the 192 MB L2, so default-temporal loads keep them resident across graph
// replays). W streams coalesced into double-buffered LDS tiles; the dot
// products run on V_WMMA_F32_16X16X4_F32 with A = x broadcast over M and
// B = 4x16 W tile. D[0][n] (lanes 0..15, acc VGPR0) = y for the 16 rows.
// ---------------------------------------------------------------------------

typedef float v2f __attribute__((ext_vector_type(2)));
typedef float v8f __attribute__((ext_vector_type(8)));

#define TILE_ROWS  128          // output rows per workgroup (8 waves x 16)
#define KC         32           // K chunk per LDS tile
#define LDS_STRIDE 34           // padded row stride (floats): conflict-free b64
#define MAX_K      5120

__global__ __launch_bounds__(256)
void gemv_wmma(const float* __restrict__ W, const float* __restrict__ x,
               const float* __restrict__ bias, float* __restrict__ y,
               int M, int K, int doRelu)
{
    __shared__ float sX[MAX_K + KC];                  // zero-padded x
    __shared__ float sW[2][TILE_ROWS * LDS_STRIDE];   // double-buffered tiles

    const int tid  = threadIdx.x;
    const int lane = tid & 31;
    const int wave = tid >> 5;
    const int n    = lane & 15;       // B column = output row within wave tile
    const int hi   = (lane >> 4) & 1; // lane half: K={0,1} vs K={2,3}
    const int rowbase = blockIdx.x * TILE_ROWS;

    const int Kpad    = (K + KC - 1) & ~(KC - 1);
    const int nchunks = Kpad / KC;

    // Stage x into LDS: branch-free main copy + tiny zero-pad tail.
    for (int k = tid; k < K; k += 256) sX[k] = x[k];
    if (tid < Kpad - K) sX[K + tid] = 0.0f;

    // Per-thread tile slice: 4 passes x 4 consecutive floats (8 thr/row).
    const int rl0 = tid >> 3;          // 0..31
    const int c0  = (tid & 7) * 4;     // 0,4,...,28
    float r[4][4];

    auto load_chunk = [&](int kb) {
        const int k = kb + c0;
        #pragma unroll
        for (int pass = 0; pass < 4; ++pass) {
            const int row = rowbase + pass * 32 + rl0;
            float v0 = 0.f, v1 = 0.f, v2 = 0.f, v3 = 0.f;
            if (row < M) {
                const float* p = W + (size_t)row * (size_t)K + k;
                if (k + 3 < K) {                  // fast path, mergeable
                    v0 = p[0]; v1 = p[1]; v2 = p[2]; v3 = p[3];
                } else {                          // K-edge guards
                    if (k + 0 < K) v0 = p[0];
                    if (k + 1 < K) v1 = p[1];
                    if (k + 2 < K) v2 = p[2];
                }
            }
            r[pass][0] = v0; r[pass][1] = v1; r[pass][2] = v2; r[pass][3] = v3;
        }
    };
    auto store_chunk = [&](int buf) {
        #pragma unroll
        for (int pass = 0; pass < 4; ++pass) {
            float* q = &sW[buf][(pass * 32 + rl0) * LDS_STRIDE + c0];
            q[0] = r[pass][0]; q[1] = r[pass][1];
            q[2] = r[pass][2]; q[3] = r[pass][3];
        }
    };

    // Prologue: chunk 0 -> buffer 0.
    load_chunk(0);
    store_chunk(0);
    __syncthreads();

    v8f c = {};                          // f32 16x16 accumulator
    const int trow = wave * 16 + n;      // tile row this lane reads for B

    for (int i = 0; i < nchunks; ++i) {
        const int kb = i * KC;
        // Issue next chunk's global loads before computing -> overlap HBM
        // latency with the WMMA + LDS phase (occupancy here is ~2 waves/SIMD).
        if (i + 1 < nchunks) load_chunk(kb + KC);

        // 8 WMMAs on the current buffer. EXEC all ones (no divergence).
        // A (16x4, x bcast over M): VGPR0 = K=(hi?2:0), VGPR1 = K=(hi?3:1)
        // B (4x16):                 VGPR0 = B[K=(hi?2:0)][n], VGPR1 = next K
        const float* wbuf = sW[i & 1];
        #pragma unroll
        for (int j = 0; j < KC / 4; ++j) {
            const int kk = j * 4 + (hi ? 2 : 0);
            v2f a, b;
            a.x = sX[kb + kk];
            a.y = sX[kb + kk + 1];
            b.x = wbuf[trow * LDS_STRIDE + kk];
            b.y = wbuf[trow * LDS_STRIDE + kk + 1];
            c = __builtin_amdgcn_wmma_f32_16x16x4_f32(
                    false, a, false, b, (short)0, c, false, false);
        }

        if (i + 1 < nchunks) {
            __syncthreads();             // everyone done reading buf (i+1)&1
            store_chunk((i + 1) & 1);
            __syncthreads();             // stores visible before next compute
        }
    }

    // D[0][n]: acc VGPR0, lanes 0..15 hold (M=0, N=0..15).
    if (lane < 16) {
        const int row = rowbase + wave * 16 + lane;
        if (row < M) {
            float v = c[0] + bias[row];
            if (doRelu) v = fmaxf(v, 0.0f);
            y[row] = v;
        }
    }
}

// concat(spatial, structural) -> a1[1950];  (self + 3 nbrs)/4 -> agg[1310]
__global__ __launch_bounds__(256)
void prep_kernel(const float* __restrict__ spatial,
                 const float* __restrict__ structural,
                 const int* __restrict__ neighbour,
                 float* __restrict__ a1, float* __restrict__ agg)
{
    const int tid = threadIdx.x;
    for (int i = tid; i < 1950; i += 256)
        a1[i] = (i < 640) ? spatial[i] : structural[i - 640];

    for (int i = tid; i < 1310; i += 256) {
        const int node = i / 131;
        const int f    = i % 131;
        int n0 = neighbour[node * 3 + 0];
        int n1 = neighbour[node * 3 + 1];
        int n2 = neighbour[node * 3 + 2];
        n0 = ((unsigned)n0 < 10u) ? n0 : 0;
        n1 = ((unsigned)n1 < 10u) ? n1 : 0;
        n2 = ((unsigned)n2 < 10u) ? n2 : 0;
        const float s = structural[node * 131 + f]
                      + structural[n0 * 131 + f]
                      + structural[n1 * 131 + f]
                      + structural[n2 * 131 + f];
        agg[i] = s * 0.25f;
    }
}

extern "C" void kernel_launch(void* const* d_in, const int* in_sizes, int n_in,
                              void* d_out, int out_size, void* d_ws, size_t ws_size,
                              hipStream_t stream)
{
    const float* spatial    = (const float*)d_in[0];
    const float* structural = (const float*)d_in[1];
    const int*   neighbour  = (const int*)  d_in[2];
    const float* W1 = (const float*)d_in[3];
    const float* b1 = (const float*)d_in[4];
    const float* W2 = (const float*)d_in[5];
    const float* b2 = (const float*)d_in[6];
    const float* W3 = (const float*)d_in[7];
    const float* b3 = (const float*)d_in[8];
    const float* W4 = (const float*)d_in[9];
    const float* b4 = (const float*)d_in[10];

    float* out = (float*)d_out;          // out1[2560] ++ out2[2560]
    float* ws  = (float*)d_ws;
    float* a1  = ws;                     // 1950
    float* agg = ws + 2048;              // 1310
    float* h1  = ws + 4096;              // 2000
    float* h2  = ws + 8192;              // 5120  (total 53 KB of ws)

    prep_kernel<<<1, 256, 0, stream>>>(spatial, structural, neighbour, a1, agg);

    // h1 = relu(W1 @ a1 + b1)         [2000 x 1950]
    gemv_wmma<<<(2000 + TILE_ROWS - 1) / TILE_ROWS, 256, 0, stream>>>(
        W1, a1, b1, h1, 2000, 1950, 1);
    // h2 = relu(W3 @ agg + b3)        [5120 x 1310]
    gemv_wmma<<<(5120 + TILE_ROWS - 1) / TILE_ROWS, 256, 0, stream>>>(
        W3, agg, b3, h2, 5120, 1310, 1);
    // out1 = W2 @ h1 + b2             [2560 x 2000]
    gemv_wmma<<<(2560 + TILE_ROWS - 1) / TILE_ROWS, 256, 0, stream>>>(
        W2, h1, b2, out, 2560, 2000, 0);
    // out2 = W4 @ h2 + b4             [2560 x 5120]
    gemv_wmma<<<(2560 + TILE_ROWS - 1) / TILE_ROWS, 256, 0, stream>>>(
        W4, h2, b4, out + 2560, 2560, 5120, 0);
}